// LightGCN_14594298871944
// MI455X (gfx1250) — compile-verified
//
#include <hip/hip_runtime.h>

// LightGCN propagation for MI455X (gfx1250, wave32).
// CSR pull-mode SpMM (no float atomics in the hot loop, accum fused) + WMMA finalize.
// Fallback: push-mode atomic scatter if workspace is too small for CSR.

typedef float v2f __attribute__((ext_vector_type(2)));
typedef float v8f __attribute__((ext_vector_type(8)));

#define EMBED 64

// ---------------- zero fill ----------------
__global__ void k_zero_f(float* __restrict__ p, size_t n) {
    size_t i = (size_t)blockIdx.x * blockDim.x + threadIdx.x;
    size_t stride = (size_t)gridDim.x * blockDim.x;
    size_t n4 = n >> 2;
    float4* p4 = (float4*)p;
    for (size_t j = i; j < n4; j += stride) p4[j] = make_float4(0.f, 0.f, 0.f, 0.f);
    for (size_t j = (n4 << 2) + i; j < n; j += stride) p[j] = 0.f;
}

__global__ void k_zero_i(int* __restrict__ p, int n) {
    int i = blockIdx.x * blockDim.x + threadIdx.x;
    int stride = gridDim.x * blockDim.x;
    for (int j = i; j < n; j += stride) p[j] = 0;
}

// ---------------- degree count (int) ----------------
__global__ void k_count(const int* __restrict__ rows, int* __restrict__ cnt, int nedges) {
    int i = blockIdx.x * blockDim.x + threadIdx.x;
    int stride = gridDim.x * blockDim.x;
    for (int e = i; e < nedges; e += stride) atomicAdd(&cnt[rows[e]], 1);
}

// ---------------- dis = deg^-1/2 ----------------
__global__ void k_dis(const int* __restrict__ cnt, float* __restrict__ dis, int n) {
    int i = blockIdx.x * blockDim.x + threadIdx.x;
    int stride = gridDim.x * blockDim.x;
    for (int j = i; j < n; j += stride) {
        int d = cnt[j];
        dis[j] = (d > 0) ? rsqrtf((float)d) : 0.f;
    }
}

// ---------------- scan stage 1: per-block sums (256 values/block) ----------------
__global__ void k_blocksum(const int* __restrict__ cnt, int* __restrict__ bsum, int n) {
    __shared__ int sm[256];
    int i = blockIdx.x * 256 + threadIdx.x;
    sm[threadIdx.x] = (i < n) ? cnt[i] : 0;
    __syncthreads();
    for (int s = 128; s > 0; s >>= 1) {
        if (threadIdx.x < s) sm[threadIdx.x] += sm[threadIdx.x + s];
        __syncthreads();
    }
    if (threadIdx.x == 0) bsum[blockIdx.x] = sm[0];
}

// ---------------- scan stage 2: serial exclusive scan of block sums (tiny) ----------------
__global__ void k_scan_top(int* __restrict__ bsum, int nb, int* __restrict__ row_start,
                           int n, int nedges) {
    if (blockIdx.x == 0 && threadIdx.x == 0) {
        int run = 0;
        for (int b = 0; b < nb; ++b) { int v = bsum[b]; bsum[b] = run; run += v; }
        row_start[n] = nedges;
    }
}

// ---------------- scan stage 3: per-block exclusive scan + block offset ----------------
__global__ void k_scan_block(const int* __restrict__ cnt, const int* __restrict__ bsum,
                             int* __restrict__ row_start, int n) {
    __shared__ int sm[256];
    int i = blockIdx.x * 256 + threadIdx.x;
    int v = (i < n) ? cnt[i] : 0;
    sm[threadIdx.x] = v;
    __syncthreads();
    for (int s = 1; s < 256; s <<= 1) {   // Hillis-Steele inclusive scan
        int t = (threadIdx.x >= s) ? sm[threadIdx.x - s] : 0;
        __syncthreads();
        sm[threadIdx.x] += t;
        __syncthreads();
    }
    if (i < n) row_start[i] = sm[threadIdx.x] - v + bsum[blockIdx.x];
}

// ---------------- CSR fill (counting sort of edges by destination) ----------------
__global__ void k_fill(const int* __restrict__ rows, const int* __restrict__ cols,
                       int* __restrict__ cursor, const int* __restrict__ row_start,
                       int* __restrict__ ecol, int nedges) {
    int i = blockIdx.x * blockDim.x + threadIdx.x;
    int stride = gridDim.x * blockDim.x;
    for (int e = i; e < nedges; e += stride) {
        int r = rows[e];
        int pos = row_start[r] + atomicAdd(&cursor[r], 1);
        ecol[pos] = cols[e];
    }
}

// ---------------- init: x_cur = emb, acc(d_out) = emb ----------------
__global__ void k_init(const float* __restrict__ emb, float* __restrict__ xcur,
                       float* __restrict__ acc, size_t n) {
    size_t i = (size_t)blockIdx.x * blockDim.x + threadIdx.x;
    size_t stride = (size_t)gridDim.x * blockDim.x;
    size_t n4 = n >> 2;
    const float4* e4 = (const float4*)emb;
    float4* x4 = (float4*)xcur;
    float4* a4 = (float4*)acc;
    for (size_t j = i; j < n4; j += stride) {
        float4 v = e4[j];
        x4[j] = v;
        a4[j] = v;
    }
    for (size_t j = (n4 << 2) + i; j < n; j += stride) {
        float v = emb[j];
        xcur[j] = v;
        acc[j] = v;
    }
}

// ---------------- pull-mode SpMM, fused accumulate: one destination row per wave32 ----
// row = dis[r] * sum_{j in row r} dis[ecol[j]] * xcur[ecol[j],:]
// acc[r,:] += row;  if (store_next) xnext[r,:] = row
// No atomics, no output zeroing; 2 dims per lane; 2-edge unroll for latency hiding.
__global__ void k_pull(const int* __restrict__ row_start, const int* __restrict__ ecol,
                       const float* __restrict__ dis, const float* __restrict__ xcur,
                       float* __restrict__ xnext, float* __restrict__ acc,
                       int n, int store_next) {
    int lane = threadIdx.x & 31;
    int wave = (blockIdx.x * blockDim.x + threadIdx.x) >> 5;
    int nwaves = (gridDim.x * blockDim.x) >> 5;
    for (int r = wave; r < n; r += nwaves) {       // wave-uniform
        int s = row_start[r];
        int t = row_start[r + 1];
        float ax = 0.f, ay = 0.f;
        int j = s;
        for (; j + 1 < t; j += 2) {                // 2 independent edges in flight
            int c0 = ecol[j];
            int c1 = ecol[j + 1];
            float w0 = dis[c0];
            float w1 = dis[c1];
            const float2 x0 = ((const float2*)(xcur + (size_t)c0 * EMBED))[lane];
            const float2 x1 = ((const float2*)(xcur + (size_t)c1 * EMBED))[lane];
            ax += w0 * x0.x;
            ay += w0 * x0.y;
            ax += w1 * x1.x;
            ay += w1 * x1.y;
        }
        if (j < t) {                               // wave-uniform tail
            int c = ecol[j];
            float w = dis[c];
            const float2 xv = ((const float2*)(xcur + (size_t)c * EMBED))[lane];
            ax += w * xv.x;
            ay += w * xv.y;
        }
        float dr = dis[r];
        float rx = dr * ax;
        float ry = dr * ay;
        float2* arow = (float2*)(acc + (size_t)r * EMBED);
        float2 prev = arow[lane];
        arow[lane] = make_float2(prev.x + rx, prev.y + ry);
        if (store_next) {                          // wave-uniform flag
            ((float2*)(xnext + (size_t)r * EMBED))[lane] = make_float2(rx, ry);
        }
    }
}

// ---------------- fallback push scatter (atomic) ----------------
__global__ void k_scatter(const int* __restrict__ rows, const int* __restrict__ cols,
                          const float* __restrict__ dis, const float* __restrict__ xcur,
                          float* __restrict__ xnext, int nedges) {
    int lane = threadIdx.x & 31;
    int wave = (blockIdx.x * blockDim.x + threadIdx.x) >> 5;
    int nwaves = (gridDim.x * blockDim.x) >> 5;
    for (int e = wave; e < nedges; e += nwaves) {
        if (e + nwaves < nedges) {
            __builtin_prefetch(&rows[e + nwaves], 0, 0);
            __builtin_prefetch(&cols[e + nwaves], 0, 0);
        }
        int r = rows[e];
        int c = cols[e];
        float nrm = dis[r] * dis[c];
        const float2 xv = ((const float2*)(xcur + (size_t)c * EMBED))[lane];
        float* dst = xnext + (size_t)r * EMBED + lane * 2;
        atomicAdd(dst,     nrm * xv.x);
        atomicAdd(dst + 1, nrm * xv.y);
    }
}

// ---------------- acc += xnext (push fallback only) ----------------
__global__ void k_accum(float* __restrict__ acc, const float* __restrict__ xn, size_t n) {
    size_t i = (size_t)blockIdx.x * blockDim.x + threadIdx.x;
    size_t stride = (size_t)gridDim.x * blockDim.x;
    size_t n4 = n >> 2;
    float4* a4 = (float4*)acc;
    const float4* x4 = (const float4*)xn;
    for (size_t j = i; j < n4; j += stride) {
        float4 a = a4[j];
        float4 x = x4[j];
        a.x += x.x; a.y += x.y; a.z += x.z; a.w += x.w;
        a4[j] = a;
    }
    for (size_t j = (n4 << 2) + i; j < n; j += stride) acc[j] += xn[j];
}

// ---------------- finalize: acc *= 0.25 via V_WMMA_F32_16X16X4_F32 ----------------
// D = sum_j A_j(16x4)*B_j(4x16) with A_j = columns 4j..4j+3 of 0.25*I16 -> exact scale.
__global__ void k_finalize_wmma(float* __restrict__ acc, int ntiles) {
    int lane = threadIdx.x & 31;
    int wave = (blockIdx.x * blockDim.x + threadIdx.x) >> 5;
    int nwaves = (gridDim.x * blockDim.x) >> 5;
    int n = lane & 15;
    int khalf = lane >> 4;
    for (int t = wave; t < ntiles; t += nwaves) {  // wave-uniform: EXEC stays full
        int nb = t >> 2;
        int db = t & 3;
        size_t base = (size_t)nb * 16 * EMBED + (size_t)db * 16;
        v8f c = {};
#pragma unroll
        for (int j = 0; j < 4; ++j) {
            int k0 = 4 * j + 2 * khalf;
            v2f a, b;
            a.x = (n == k0)     ? 0.25f : 0.0f;
            a.y = (n == k0 + 1) ? 0.25f : 0.0f;
            b.x = acc[base + (size_t)k0 * EMBED + n];
            b.y = acc[base + (size_t)(k0 + 1) * EMBED + n];
            c = __builtin_amdgcn_wmma_f32_16x16x4_f32(
                    false, a, false, b, (short)0, c, false, false);
        }
#pragma unroll
        for (int r = 0; r < 8; ++r) {
            acc[base + (size_t)(r + 8 * khalf) * EMBED + n] = c[r];
        }
    }
}

__global__ void k_scale_tail(float* __restrict__ acc, size_t start, size_t n) {
    size_t i = start + (size_t)blockIdx.x * blockDim.x + threadIdx.x;
    size_t stride = (size_t)gridDim.x * blockDim.x;
    for (size_t j = i; j < n; j += stride) acc[j] *= 0.25f;
}

static inline int grid_for(size_t work, int block, int cap) {
    size_t g = (work + (size_t)block - 1) / (size_t)block;
    if (g > (size_t)cap) g = (size_t)cap;
    if (g < 1) g = 1;
    return (int)g;
}

extern "C" void kernel_launch(void* const* d_in, const int* in_sizes, int n_in,
                              void* d_out, int out_size, void* d_ws, size_t ws_size,
                              hipStream_t stream) {
    const float* emb = (const float*)d_in[0];
    const int* ei = (const int*)d_in[1];

    const int N = in_sizes[0] / EMBED;        // 256000
    const int E = in_sizes[1] / 2;            // 4000000
    const int* rows = ei;
    const int* cols = ei + E;
    const size_t feat = (size_t)N * EMBED;
    const int B = 256;
    const int NB = (N + 255) / 256;           // scan blocks

    float* acc = (float*)d_out;

    // CSR workspace layout (4-byte units, all offsets multiple of 4 elems):
    //   cnt/cursor[N] | dis[N] | row_start[N+64] | bsum[pad 4096] | ecol[E] | xa | xb
    size_t csr_words = (size_t)N + N + (N + 64) + 4096 + E + 2 * feat;
    bool use_csr = (ws_size >= csr_words * sizeof(float));

    if (use_csr) {
        int*   cnt    = (int*)d_ws;           // later reused as cursor
        float* dis    = (float*)(cnt + N);
        int*   rstart = (int*)(dis + N);
        int*   bsum   = rstart + N + 64;
        int*   ecol   = bsum + 4096;
        float* xa     = (float*)(ecol + E);
        float* xb     = xa + feat;

        // degrees + dis
        k_zero_i<<<grid_for(N, B, 4096), B, 0, stream>>>(cnt, N);
        k_count<<<grid_for(E, B, 8192), B, 0, stream>>>(rows, cnt, E);
        k_dis<<<grid_for(N, B, 4096), B, 0, stream>>>(cnt, dis, N);

        // exclusive scan -> row_start
        k_blocksum<<<NB, 256, 0, stream>>>(cnt, bsum, N);
        k_scan_top<<<1, 1, 0, stream>>>(bsum, NB, rstart, N, E);
        k_scan_block<<<NB, 256, 0, stream>>>(cnt, bsum, rstart, N);

        // counting sort edges by destination (cnt reused as cursor)
        k_zero_i<<<grid_for(N, B, 4096), B, 0, stream>>>(cnt, N);
        k_fill<<<grid_for(E, B, 8192), B, 0, stream>>>(rows, cols, cnt, rstart, ecol, E);

        // init + 3 fused pull layers (no atomics, no zeroing, no separate accum)
        k_init<<<grid_for(feat / 4, B, 8192), B, 0, stream>>>(emb, xa, acc, feat);
        float* xcur = xa;
        float* xnext = xb;
        for (int l = 0; l < 3; ++l) {
            int store_next = (l < 2) ? 1 : 0;  // last layer feeds only acc
            k_pull<<<grid_for((size_t)N * 32, B, 8192), B, 0, stream>>>(
                rstart, ecol, dis, xcur, xnext, acc, N, store_next);
            float* t = xcur; xcur = xnext; xnext = t;
        }
    } else {
        // fallback: push-mode atomic scatter (smaller workspace: ~133 MB)
        int*   cnt = (int*)d_ws;
        float* dis = (float*)(cnt + N);
        float* xa  = dis + N;
        float* xb  = xa + feat;

        k_zero_i<<<grid_for(N, B, 4096), B, 0, stream>>>(cnt, N);
        k_count<<<grid_for(E, B, 8192), B, 0, stream>>>(rows, cnt, E);
        k_dis<<<grid_for(N, B, 4096), B, 0, stream>>>(cnt, dis, N);

        k_init<<<grid_for(feat / 4, B, 8192), B, 0, stream>>>(emb, xa, acc, feat);
        float* xcur = xa;
        float* xnext = xb;
        for (int l = 0; l < 3; ++l) {
            k_zero_f<<<grid_for(feat / 4, B, 8192), B, 0, stream>>>(xnext, feat);
            k_scatter<<<4096, B, 0, stream>>>(rows, cols, dis, xcur, xnext, E);
            k_accum<<<grid_for(feat / 4, B, 8192), B, 0, stream>>>(acc, xnext, feat);
            float* t = xcur; xcur = xnext; xnext = t;
        }
    }

    // finalize acc *= 0.25 with WMMA tiles; scalar tail for N % 16 != 0
    int full_nodes = N & ~15;
    int ntiles = (full_nodes / 16) * (EMBED / 16);
    if (ntiles > 0) {
        k_finalize_wmma<<<grid_for((size_t)ntiles * 32, B, 2048), B, 0, stream>>>(acc, ntiles);
    }
    if (full_nodes < N) {
        size_t start = (size_t)full_nodes * EMBED;
        k_scale_tail<<<grid_for(feat - start, B, 256), B, 0, stream>>>(acc, start, feat);
    }
}